// GATNet_61014305407688
// MI455X (gfx1250) — compile-verified
//
#include <hip/hip_runtime.h>
#include <hip/hip_bf16.h>

// ---------------- problem constants (match reference) ----------------
#define NN        50000
#define EE        800000      // edges before self loops
#define ETOT      (EE + NN)   // with self loops
#define GG        64
#define HEADS     4
#define HID       25
#define DENSE     100         // HEADS*HID
#define IN_DIM    128
#define OUT_DIM   2
#define NUM_PROT  10
#define EPS_      1e-4f
#define NEG_SLOPE 0.2f

#define COLS_PAD  112         // DENSE padded to 7 tiles of 16
#define TILES_N   7
#define KPAD_MAX  (IN_DIM + 2)

typedef float v2f __attribute__((ext_vector_type(2)));
typedef float v8f __attribute__((ext_vector_type(8)));

// order-preserving encode of float into uint so unsigned atomicMax == float max
__device__ __forceinline__ unsigned fenc(float f) {
  unsigned b = __float_as_uint(f);
  return (b & 0x80000000u) ? ~b : (b | 0x80000000u);
}
__device__ __forceinline__ float fdec(unsigned e) {
  unsigned b = (e & 0x80000000u) ? (e & 0x7FFFFFFFu) : ~e;
  return __uint_as_float(b);
}
#define ENC_NEG_INF 0x007FFFFFu   // fenc(-inf)

// ---------------- generic fills ----------------
__global__ void k_fill_f32(float* p, float v, int n) {
  int i = blockIdx.x * blockDim.x + threadIdx.x;
  if (i < n) p[i] = v;
}
__global__ void k_fill_u32(unsigned* p, unsigned v, int n) {
  int i = blockIdx.x * blockDim.x + threadIdx.x;
  if (i < n) p[i] = v;
}

// ---------------- fp32 WMMA GEMM: out[N,100] = A[N,K] @ W[K,100] ----------------
// 256 threads = 8 waves; each wave computes a 16-row x 112-col stripe.
// K in {128,100}, both % 4 == 0. W staged TRANSPOSED in LDS: Wlds[c*KPAD + k],
// KPAD = K+2 (even -> v2f alignment; stride 2 banks mod 64 -> conflict-free).
// Inner loop: batch 7 DS loads into distinct regs, prefetch next A, then 7 WMMAs.
__global__ __launch_bounds__(256)
void k_gemm_wmma(const float* __restrict__ A, const float* __restrict__ W,
                 float* __restrict__ out, int Nrows, int K) {
  __shared__ float Wlds[COLS_PAD * KPAD_MAX];   // 112*130 floats = 58,240 B
  const int KPAD = K + 2;
  const int tid = threadIdx.x;
  for (int i = tid; i < K * COLS_PAD; i += 256) {
    int k = i / COLS_PAD, c = i - k * COLS_PAD;
    Wlds[c * KPAD + k] = (c < DENSE) ? W[k * DENSE + c] : 0.0f;
  }
  __syncthreads();

  const int wave = tid >> 5;
  const int lane = tid & 31;
  const int half = lane >> 4;    // 0: lanes 0-15, 1: lanes 16-31
  const int l16  = lane & 15;
  const int rowBase = blockIdx.x * 128 + wave * 16;
  const int aRow = rowBase + l16;
  // Clamp instead of predicating: out-of-range A rows only feed masked-out D rows.
  const int aRowC = (aRow < Nrows) ? aRow : (Nrows - 1);
  const float* __restrict__ Aptr = A + (long)aRowC * K + 2 * half;
  const float* __restrict__ Bbase = Wlds + (long)l16 * KPAD + 2 * half;

  v8f acc[TILES_N] = {};

  v2f a = *(const v2f*)(Aptr);           // prefetch first A fragment
  for (int k0 = 0; k0 < K; k0 += 4) {
    // batch-load all 7 B fragments into distinct registers (one s_wait, not 7)
    v2f b[TILES_N];
#pragma unroll
    for (int t = 0; t < TILES_N; ++t)
      b[t] = *(const v2f*)(Bbase + (long)t * 16 * KPAD + k0);
    // prefetch next A fragment so global latency hides under the WMMA chain
    const v2f aCur = a;
    if (k0 + 4 < K) a = *(const v2f*)(Aptr + k0 + 4);
#pragma unroll
    for (int t = 0; t < TILES_N; ++t) {
      // D = A(16x4, f32) * B(4x16, f32) + C  -- exact fp32, matches reference math
      acc[t] = __builtin_amdgcn_wmma_f32_16x16x4_f32(
          false, aCur, false, b[t], (short)0, acc[t], false, false);
    }
  }

  // C/D layout: VGPR i -> row = rowBase + i + 8*half, col = t*16 + l16
#pragma unroll
  for (int t = 0; t < TILES_N; ++t) {
    const int col = t * 16 + l16;
    if (col < DENSE) {
#pragma unroll
      for (int i = 0; i < 8; ++i) {
        const int row = rowBase + i + 8 * half;
        if (row < Nrows) out[(long)row * DENSE + col] = acc[t][i];
      }
    }
  }
}

// ---------------- per-node attention coefficients + state init ----------------
__global__ void k_node_e(const float* __restrict__ hW,
                         const float* __restrict__ a_src,
                         const float* __restrict__ a_dst,
                         float* __restrict__ e_src, float* __restrict__ e_dst,
                         unsigned* __restrict__ amax, float* __restrict__ denom) {
  int i = blockIdx.x * blockDim.x + threadIdx.x;   // i = n*HEADS + h
  if (i >= NN * HEADS) return;
  int n = i >> 2, h = i & 3;
  const float* hp  = hW + (long)n * DENSE + h * HID;
  const float* asp = a_src + h * HID;
  const float* adp = a_dst + h * HID;
  float s = 0.f, d = 0.f;
#pragma unroll
  for (int j = 0; j < HID; ++j) { float v = hp[j]; s += v * asp[j]; d += v * adp[j]; }
  e_src[i] = s;
  e_dst[i] = d;
  amax[i]  = ENC_NEG_INF;
  denom[i] = 0.0f;
}

__device__ __forceinline__ void edge_nodes(const int* __restrict__ ei, int e, int& s, int& d) {
  if (e < EE) { s = ei[e]; d = ei[EE + e]; }
  else        { s = d = e - EE; }            // self loop
}

// ---------------- edge pass 1: segment max of leaky-relu'd alpha over dst ----------------
__global__ void k_edge_max(const int* __restrict__ ei,
                           const float* __restrict__ e_src, const float* __restrict__ e_dst,
                           unsigned* __restrict__ amax) {
  int e = blockIdx.x * blockDim.x + threadIdx.x;
  if (e >= ETOT) return;
  int s, d; edge_nodes(ei, e, s, d);
#pragma unroll
  for (int h = 0; h < HEADS; ++h) {
    float a = e_src[s * HEADS + h] + e_dst[d * HEADS + h];
    a = (a > 0.f) ? a : NEG_SLOPE * a;
    atomicMax(&amax[d * HEADS + h], fenc(a));
  }
}

// ---------------- edge pass 2: exp + segment sum ----------------
__global__ void k_edge_exp(const int* __restrict__ ei,
                           const float* __restrict__ e_src, const float* __restrict__ e_dst,
                           const unsigned* __restrict__ amax,
                           float* __restrict__ denom, float* __restrict__ ex) {
  int e = blockIdx.x * blockDim.x + threadIdx.x;
  if (e >= ETOT) return;
  int s, d; edge_nodes(ei, e, s, d);
#pragma unroll
  for (int h = 0; h < HEADS; ++h) {
    float a = e_src[s * HEADS + h] + e_dst[d * HEADS + h];
    a = (a > 0.f) ? a : NEG_SLOPE * a;
    float v = expf(a - fdec(amax[d * HEADS + h]));
    ex[(long)e * HEADS + h] = v;
    atomicAdd(&denom[d * HEADS + h], v);
  }
}

// ---------------- edge pass 3: weighted aggregation msg = h[src]*a -> acc[dst] ----------------
__global__ void k_edge_agg(const int* __restrict__ ei,
                           const float* __restrict__ hW,
                           const float* __restrict__ ex, const float* __restrict__ denom,
                           float* __restrict__ acc) {
  int idx = blockIdx.x * blockDim.x + threadIdx.x;   // over ETOT*DENSE = 85M < 2^31
  if (idx >= ETOT * DENSE) return;
  int e = idx / DENSE;
  int f = idx - e * DENSE;
  int h = f / HID;
  int s, d; edge_nodes(ei, e, s, d);
  float a = ex[(long)e * HEADS + h] / (denom[d * HEADS + h] + 1e-16f);
  atomicAdd(&acc[(long)d * DENSE + f], hW[(long)s * DENSE + f] * a);
}

// ---------------- bias + relu, in place ----------------
__global__ void k_finalize(float* __restrict__ acc, const float* __restrict__ bias) {
  int i = blockIdx.x * blockDim.x + threadIdx.x;
  if (i >= NN * DENSE) return;
  int f = i % DENSE;
  acc[i] = fmaxf(acc[i] + bias[f], 0.0f);
}

// ---------------- global max pool over batch ----------------
__global__ void k_pool_max(const float* __restrict__ node_emb, const int* __restrict__ batch,
                           unsigned* __restrict__ poolenc) {
  int i = blockIdx.x * blockDim.x + threadIdx.x;
  if (i >= NN * DENSE) return;
  int n = i / DENSE;
  int f = i - n * DENSE;
  atomicMax(&poolenc[batch[n] * DENSE + f], fenc(node_emb[i]));
}
__global__ void k_pool_write(const unsigned* __restrict__ poolenc, float* __restrict__ ge) {
  int i = blockIdx.x * blockDim.x + threadIdx.x;
  if (i < GG * DENSE) ge[i] = fdec(poolenc[i]);
}

// ---------------- prototype head: dist, sim, logits, softmax (single block) ----------------
__global__ __launch_bounds__(640)
void k_head(const float* __restrict__ ge, const float* __restrict__ protos,
            const float* __restrict__ lw,
            float* __restrict__ logits, float* __restrict__ probs, float* __restrict__ dist) {
  __shared__ float sge[GG * DENSE];       // 25.6 KB
  __shared__ float sg2[GG];
  __shared__ float sp2[NUM_PROT];
  __shared__ float ssim[GG * NUM_PROT];
  int t = threadIdx.x;
  for (int i = t; i < GG * DENSE; i += blockDim.x) sge[i] = ge[i];
  __syncthreads();
  if (t < GG) {
    float s = 0.f;
    for (int j = 0; j < DENSE; ++j) { float v = sge[t * DENSE + j]; s += v * v; }
    sg2[t] = s;
  } else if (t < GG + NUM_PROT) {
    int p = t - GG;
    float s = 0.f;
    for (int j = 0; j < DENSE; ++j) { float v = protos[p * DENSE + j]; s += v * v; }
    sp2[p] = s;
  }
  __syncthreads();
  if (t < GG * NUM_PROT) {
    int g = t / NUM_PROT, p = t - g * NUM_PROT;
    float xp = 0.f;
    for (int j = 0; j < DENSE; ++j) xp += sge[g * DENSE + j] * protos[p * DENSE + j];
    float dd = sg2[g] + sp2[p] - 2.0f * xp;
    dist[t] = dd;
    ssim[t] = logf((dd + 1.0f) / (dd + EPS_));
  }
  __syncthreads();
  if (t < GG) {
    float l0 = 0.f, l1 = 0.f;
#pragma unroll
    for (int p = 0; p < NUM_PROT; ++p) {
      float s = ssim[t * NUM_PROT + p];
      l0 += s * lw[p];
      l1 += s * lw[NUM_PROT + p];
    }
    logits[t * 2] = l0; logits[t * 2 + 1] = l1;
    float m = fmaxf(l0, l1);
    float e0 = expf(l0 - m), e1 = expf(l1 - m), s = e0 + e1;
    probs[t * 2] = e0 / s; probs[t * 2 + 1] = e1 / s;
  }
}

// ---------------- launcher ----------------
extern "C" void kernel_launch(void* const* d_in, const int* in_sizes, int n_in,
                              void* d_out, int out_size, void* d_ws, size_t ws_size,
                              hipStream_t stream) {
  const float* x     = (const float*)d_in[0];
  const int*   ei    = (const int*)d_in[1];     // [2,E] flat: [0..E)=src, [E..2E)=dst
  const int*   batch = (const int*)d_in[2];
  const float* W[3]  = {(const float*)d_in[3], (const float*)d_in[7],  (const float*)d_in[11]};
  const float* AS[3] = {(const float*)d_in[4], (const float*)d_in[8],  (const float*)d_in[12]};
  const float* AD[3] = {(const float*)d_in[5], (const float*)d_in[9],  (const float*)d_in[13]};
  const float* B_[3] = {(const float*)d_in[6], (const float*)d_in[10], (const float*)d_in[14]};
  const float* protos = (const float*)d_in[15];
  const float* last_w = (const float*)d_in[16];

  float* out       = (float*)d_out;
  float* logits    = out;                              // [64,2]
  float* probs     = out + 128;                        // [64,2]
  float* node_emb  = out + 256;                        // [N,100] -- doubles as per-layer acc
  float* graph_emb = out + 256 + (long)NN * DENSE;     // [64,100]
  float* dist      = graph_emb + GG * DENSE;           // [64,10]

  // workspace layout (floats)
  float*    ws      = (float*)d_ws;
  float*    hW      = ws;                                   // N*100
  float*    e_src   = hW + (long)NN * DENSE;                // N*4
  float*    e_dst   = e_src + NN * HEADS;                   // N*4
  unsigned* amax    = (unsigned*)(e_dst + NN * HEADS);      // N*4
  float*    denom   = (float*)(amax + NN * HEADS);          // N*4
  float*    ex      = denom + NN * HEADS;                   // ETOT*4
  unsigned* poolenc = (unsigned*)(ex + (long)ETOT * HEADS); // G*100

  const int T = 256;
  const int gGemm  = (NN + 127) / 128;
  const int gNH    = (NN * HEADS + T - 1) / T;
  const int gE     = (ETOT + T - 1) / T;
  const int gEF    = (ETOT * DENSE + T - 1) / T;
  const int gNF    = (NN * DENSE + T - 1) / T;
  const int gGF    = (GG * DENSE + T - 1) / T;

  for (int l = 0; l < 3; ++l) {
    const float* in = (l == 0) ? x : node_emb;
    const int K = (l == 0) ? IN_DIM : DENSE;

    // h = in @ W  (fp32 WMMA)
    k_gemm_wmma<<<gGemm, T, 0, stream>>>(in, W[l], hW, NN, K);
    // e_src/e_dst + init amax/denom
    k_node_e<<<gNH, T, 0, stream>>>(hW, AS[l], AD[l], e_src, e_dst, amax, denom);
    // acc := 0 (node_emb region is free now; GEMM already consumed it)
    k_fill_f32<<<gNF, T, 0, stream>>>(node_emb, 0.0f, NN * DENSE);
    // segmented softmax over incoming edges
    k_edge_max<<<gE, T, 0, stream>>>(ei, e_src, e_dst, amax);
    k_edge_exp<<<gE, T, 0, stream>>>(ei, e_src, e_dst, amax, denom, ex);
    // weighted aggregation
    k_edge_agg<<<gEF, T, 0, stream>>>(ei, hW, ex, denom, node_emb);
    // + bias, relu (in place)
    k_finalize<<<gNF, T, 0, stream>>>(node_emb, B_[l]);
  }

  // global max pool
  k_fill_u32<<<gGF, T, 0, stream>>>(poolenc, ENC_NEG_INF, GG * DENSE);
  k_pool_max<<<gNF, T, 0, stream>>>(node_emb, batch, poolenc);
  k_pool_write<<<gGF, T, 0, stream>>>(poolenc, graph_emb);

  // prototype head
  k_head<<<1, 640, 0, stream>>>(graph_emb, protos, last_w, logits, probs, dist);
}